// SumPool_38869454029096
// MI455X (gfx1250) — compile-verified
//
#include <hip/hip_runtime.h>
#include <math.h>

typedef __attribute__((ext_vector_type(2))) float v2f;
typedef __attribute__((ext_vector_type(8))) float v8f;

#define N_NODES   100000
#define N_EDGES   3200000
#define N_GRAPHS  64
#define TOTAL_LATENT 97
#define PK        100          // padded K / pooled row stride
#define OUT_DIM   128

// branch-free tanh: 1 - 2/(1+exp(2x)); v_exp_f32 + v_rcp_f32, saturates correctly
__device__ __forceinline__ float fast_tanh(float x) {
    float e = __builtin_amdgcn_exp2f(x * 2.8853900817779268f);   // exp(2x) = 2^(2*log2e*x)
    return 1.0f - 2.0f * __builtin_amdgcn_rcpf(1.0f + e);
}

// ---------------- degree computation: node_degs = (#incoming edges) + 1 ----
__global__ void init_degs_kernel(float* __restrict__ degs) {
    int i = blockIdx.x * 256 + threadIdx.x;
    if (i < N_NODES) degs[i] = 1.0f;
}

__global__ void count_degs_kernel(const int* __restrict__ dst, float* __restrict__ degs) {
    int e = blockIdx.x * 256 + threadIdx.x;
    if (e < N_EDGES) atomicAdd(&degs[dst[e]], 1.0f);
}

// degs[i] <- 1/degs[i]  (hoists 12.8M per-element divisions down to 100K)
__global__ void recip_degs_kernel(float* __restrict__ degs) {
    int i = blockIdx.x * 256 + threadIdx.x;
    if (i < N_NODES) degs[i] = 1.0f / degs[i];
}

// ---------------- agg = h (self term), vectorized ---------------------------
__global__ void copy4_kernel(const float4* __restrict__ h, float4* __restrict__ agg, int n4) {
    int i = blockIdx.x * 256 + threadIdx.x;
    if (i < n4) agg[i] = h[i];
}

// ---------------- agg[dst] += h[src] (one wave per edge, lane = feature) ---
__global__ void edge_agg_kernel(const int* __restrict__ src, const int* __restrict__ dst,
                                const float* __restrict__ h, float* __restrict__ agg) {
    int idx = blockIdx.x * 256 + threadIdx.x;
    int e = idx >> 5;
    int f = idx & 31;
    if (e < N_EDGES) {
        int s = src[e];
        int d = dst[e];
        atomicAdd(&agg[d * 32 + f], h[s * 32 + f]);   // coalesced 128B gather, L2 atomic
    }
}

// ---- layers 0..2: h_out = tanh((agg @ W[32,32] + b) * invdeg), WMMA f32 ----
// W and bias staged in LDS (padded stride 33 to avoid half-wave bank clash).
__global__ void layer32_wmma_kernel(const float* __restrict__ agg, const float* __restrict__ W,
                                    const float* __restrict__ bias, const float* __restrict__ invdeg,
                                    float* __restrict__ hout) {
    __shared__ float Wl[32 * 33];
    __shared__ float bl[32];
    for (int t = threadIdx.x; t < 1024; t += 256) {
        int r = t >> 5, cc = t & 31;
        Wl[r * 33 + cc] = W[t];
    }
    if (threadIdx.x < 32) bl[threadIdx.x] = bias[threadIdx.x];
    __syncthreads();

    int warp = (blockIdx.x * blockDim.x + threadIdx.x) >> 5;
    const int ntiles = (N_NODES / 16) * 2;            // 2 column tiles of 16
    if (warp >= ntiles) return;                        // wave-uniform exit
    int lane  = threadIdx.x & 31;
    int mtile = warp >> 1;
    int ntile = warp & 1;
    int row0  = mtile * 16;
    int mrow  = lane & 15;
    int khalf = (lane < 16) ? 0 : 2;                   // K split across half-waves
    int ncol  = ntile * 16 + mrow;

    const float* arow = agg + (size_t)(row0 + mrow) * 32 + khalf;
    v8f c = {};
    #pragma unroll
    for (int k0 = 0; k0 < 32; k0 += 4) {
        v2f a, b;
        a.x = arow[k0];
        a.y = arow[k0 + 1];
        b.x = Wl[(k0 + khalf)     * 33 + ncol];
        b.y = Wl[(k0 + khalf + 1) * 33 + ncol];
        c = __builtin_amdgcn_wmma_f32_16x16x4_f32(false, a, false, b, (short)0, c, false, false);
    }

    float bv = bl[ncol];
    int mbase = row0 + ((lane < 16) ? 0 : 8);
    #pragma unroll
    for (int i = 0; i < 8; ++i) {
        int row = mbase + i;
        hout[(size_t)row * 32 + ncol] = fast_tanh((c[i] + bv) * invdeg[row]);
    }
}

// ---- layer 3 (dout=1): wave-per-node dot product + shuffle reduction ------
__global__ void layer3_kernel(const float* __restrict__ agg, const float* __restrict__ W3,
                              const float* __restrict__ b3, const float* __restrict__ invdeg,
                              float* __restrict__ h3) {
    int idx  = blockIdx.x * 256 + threadIdx.x;
    int node = idx >> 5;
    int lane = idx & 31;
    if (node >= N_NODES) return;
    float v = agg[(size_t)node * 32 + lane] * W3[lane];
    #pragma unroll
    for (int off = 16; off > 0; off >>= 1)
        v += __shfl_down(v, off, 32);
    if (lane == 0) h3[node] = fast_tanh((v + b3[0]) * invdeg[node]);
}

// ---------------- zero init ------------------------------------------------
__global__ void zero_kernel(float* __restrict__ p, int n) {
    int i = blockIdx.x * 256 + threadIdx.x;
    if (i < n) p[i] = 0.0f;
}

// ---------------- Wout -> zero-padded [PK, OUT_DIM] ------------------------
__global__ void pad_wout_kernel(const float* __restrict__ Wout, float* __restrict__ Wpad) {
    int idx = blockIdx.x * 256 + threadIdx.x;
    if (idx < PK * OUT_DIM) {
        int row = idx >> 7;           // /128
        Wpad[idx] = (row < TOTAL_LATENT) ? Wout[idx] : 0.0f;
    }
}

// ---------------- per-graph sum pooling into pooled[64, PK] ----------------
__global__ void pool_kernel(const float* __restrict__ h, const int* __restrict__ gid,
                            float* __restrict__ pooled, int nelem, int shift, int mask, int col_off) {
    int idx = blockIdx.x * 256 + threadIdx.x;
    if (idx < nelem) {
        int node = idx >> shift;
        int f = idx & mask;
        atomicAdd(&pooled[gid[node] * PK + col_off + f], h[idx]);   // 25KB target, L2-resident
    }
}

// ---------------- out = relu(pooled @ Woutpad + bout), WMMA f32 16x16x4 ----
// pooled has stride PK=100 with zero pad cols, Woutpad is [100,128] zero-padded:
// K loop is fully unconditional.
__global__ void out_wmma_kernel(const float* __restrict__ pooled, const float* __restrict__ Wpad,
                                const float* __restrict__ bout, float* __restrict__ out) {
    int warp = (blockIdx.x * blockDim.x + threadIdx.x) >> 5;
    int lane = threadIdx.x & 31;
    if (warp >= 32) return;                // 4 row tiles x 8 col tiles, wave-uniform exit
    int mtile = warp >> 3;
    int ntile = warp & 7;
    int row0  = mtile * 16;
    int mrow  = lane & 15;
    int khalf = (lane < 16) ? 0 : 2;
    int ncol  = ntile * 16 + mrow;

    const float* arow = pooled + (row0 + mrow) * PK + khalf;
    v8f c = {};
    #pragma unroll
    for (int k0 = 0; k0 < PK; k0 += 4) {
        v2f a, b;
        a.x = arow[k0];
        a.y = arow[k0 + 1];
        b.x = Wpad[(k0 + khalf)     * OUT_DIM + ncol];
        b.y = Wpad[(k0 + khalf + 1) * OUT_DIM + ncol];
        c = __builtin_amdgcn_wmma_f32_16x16x4_f32(false, a, false, b, (short)0, c, false, false);
    }

    float bv = bout[ncol];
    int mbase = row0 + ((lane < 16) ? 0 : 8);
    #pragma unroll
    for (int i = 0; i < 8; ++i) {
        float v = c[i] + bv;
        out[(mbase + i) * OUT_DIM + ncol] = (v > 0.0f) ? v : 0.0f;
    }
}

// ---------------------------------------------------------------------------
extern "C" void kernel_launch(void* const* d_in, const int* in_sizes, int n_in,
                              void* d_out, int out_size, void* d_ws, size_t ws_size,
                              hipStream_t stream) {
    const float* node_feat = (const float*)d_in[0];
    const int*   src       = (const int*)d_in[1];
    const int*   dst       = (const int*)d_in[2];
    const int*   gid       = (const int*)d_in[3];
    // d_in[4] = num_graphs scalar (compile-time 64)
    const float* W[4] = {(const float*)d_in[5], (const float*)d_in[7],
                         (const float*)d_in[9], (const float*)d_in[11]};
    const float* B[4] = {(const float*)d_in[6], (const float*)d_in[8],
                         (const float*)d_in[10], (const float*)d_in[12]};
    const float* Wout = (const float*)d_in[13];
    const float* bout = (const float*)d_in[14];
    float* out = (float*)d_out;

    // workspace carve-up (floats)
    float* ws     = (float*)d_ws;
    float* degs   = ws;                               // N (becomes 1/deg after recip)
    float* agg    = degs + N_NODES;                   // N*32
    float* h0     = agg  + (size_t)N_NODES * 32;      // N*32
    float* h1     = h0   + (size_t)N_NODES * 32;      // N*32
    float* h2     = h1   + (size_t)N_NODES * 32;      // N*32
    float* h3     = h2   + (size_t)N_NODES * 32;      // N
    float* pooled = h3   + N_NODES;                   // 64*PK (pad cols stay zero)
    float* wpad   = pooled + N_GRAPHS * PK;           // PK*128

    // degrees -> reciprocal degrees
    init_degs_kernel<<<(N_NODES + 255) / 256, 256, 0, stream>>>(degs);
    count_degs_kernel<<<(N_EDGES + 255) / 256, 256, 0, stream>>>(dst, degs);
    recip_degs_kernel<<<(N_NODES + 255) / 256, 256, 0, stream>>>(degs);

    // GNN layers 0..2 (32 -> 32)
    const float* hin = node_feat;
    float* hs[3] = {h0, h1, h2};
    const int n4 = N_NODES * 32 / 4;
    for (int l = 0; l < 3; ++l) {
        copy4_kernel<<<(n4 + 255) / 256, 256, 0, stream>>>(
            (const float4*)hin, (float4*)agg, n4);
        edge_agg_kernel<<<(N_EDGES * 32 + 255) / 256, 256, 0, stream>>>(src, dst, hin, agg);
        int ntiles = (N_NODES / 16) * 2;
        layer32_wmma_kernel<<<(ntiles + 7) / 8, 256, 0, stream>>>(agg, W[l], B[l], degs, hs[l]);
        hin = hs[l];
    }
    // layer 3 (32 -> 1)
    copy4_kernel<<<(n4 + 255) / 256, 256, 0, stream>>>((const float4*)hin, (float4*)agg, n4);
    edge_agg_kernel<<<(N_EDGES * 32 + 255) / 256, 256, 0, stream>>>(src, dst, hin, agg);
    layer3_kernel<<<(N_NODES * 32 + 255) / 256, 256, 0, stream>>>(agg, W[3], B[3], degs, h3);

    // graph pooling of concat(h0,h1,h2,h3) -> pooled[64, PK]
    zero_kernel<<<(N_GRAPHS * PK + 255) / 256, 256, 0, stream>>>(pooled, N_GRAPHS * PK);
    pool_kernel<<<(N_NODES * 32 + 255) / 256, 256, 0, stream>>>(h0, gid, pooled, N_NODES * 32, 5, 31, 0);
    pool_kernel<<<(N_NODES * 32 + 255) / 256, 256, 0, stream>>>(h1, gid, pooled, N_NODES * 32, 5, 31, 32);
    pool_kernel<<<(N_NODES * 32 + 255) / 256, 256, 0, stream>>>(h2, gid, pooled, N_NODES * 32, 5, 31, 64);
    pool_kernel<<<(N_NODES + 255) / 256, 256, 0, stream>>>(h3, gid, pooled, N_NODES, 0, 0, 96);

    // final dense: out[64,128] = relu(pooled @ Wout + bout), padded K
    pad_wout_kernel<<<(PK * OUT_DIM + 255) / 256, 256, 0, stream>>>(Wout, wpad);
    out_wmma_kernel<<<4, 256, 0, stream>>>(pooled, wpad, bout, out);
}